// AutoEncoder_77592879170187
// MI455X (gfx1250) — compile-verified
//
#include <hip/hip_runtime.h>

// densitySmoothnessVolume for MI455X (gfx1250, wave32).
// Pipeline: zero grid -> scatter-add (f32 atomics, grid L2-resident) ->
// float4 finite-difference sweep with WMMA f32 wave reduction -> finalize.

typedef __attribute__((ext_vector_type(2))) float v2f;
typedef __attribute__((ext_vector_type(8))) float v8f;

#define XDIM 128
#define X3   (XDIM * XDIM * XDIM)

// Exact f32 full-wave32 sum using V_WMMA_F32_16X16X4_F32.
// A[m][0]=x(lane m), A[m][1]=0, A[m][2]=x(lane m+16), A[m][3]=0; B = ones.
// => D[m][n] = x(m) + x(m+16) for every n. Sum 8 D VGPRs per lane, then
// shfl_xor(16) combines the two row halves -> every lane holds the wave total.
__device__ __forceinline__ float wave_sum32(float x) {
  v2f a; a[0] = x;    a[1] = 0.0f;
  v2f b; b[0] = 1.0f; b[1] = 1.0f;
  v8f c = {};
  c = __builtin_amdgcn_wmma_f32_16x16x4_f32(
      /*neg_a=*/false, a, /*neg_b=*/false, b,
      /*c_mod=*/(short)0, c, /*reuse_a=*/false, /*reuse_b=*/false);
  float s = c[0] + c[1] + c[2] + c[3] + c[4] + c[5] + c[6] + c[7];
  s += __shfl_xor(s, 16, 32);
  return s;
}

// ---------------------------------------------------------------- zero grid
__global__ void dsv_zero_kernel(float4* __restrict__ g, float* __restrict__ acc,
                                int nAcc) {
  size_t i = (size_t)blockIdx.x * blockDim.x + threadIdx.x;
  g[i] = make_float4(0.f, 0.f, 0.f, 0.f);
  if (blockIdx.x == 0 && (int)threadIdx.x < nAcc) acc[threadIdx.x] = 0.f;
}

// ---------------------------------------------------------------- scatter
// One thread per index i: compute flat voxel once, loop over batches.
// vals[b*N + i] reads are lane-coalesced; atomics land in the L2-resident grid.
__global__ void dsv_scatter_kernel(const int* __restrict__ idx,
                                   const float* __restrict__ vals,
                                   float* __restrict__ grid, int N, int B) {
  int i = blockIdx.x * blockDim.x + threadIdx.x;
  if (i >= N) return;
  int z = idx[3 * i + 0];
  int y = idx[3 * i + 1];
  int x = idx[3 * i + 2];
  int f = (z * XDIM + y) * XDIM + x;
#pragma unroll 4
  for (int b = 0; b < B; ++b) {
    atomicAdd(grid + (size_t)b * X3 + f, vals[(size_t)b * N + i]);
  }
}

// ---------------------------------------------------------------- reduction
// gridDim.y = batch. Each thread handles `iters` float4 chunks (4 voxels),
// computing +x diffs in-register and +y/+z diffs via coalesced float4 loads.
__global__ __launch_bounds__(256)
void dsv_reduce_kernel(const float* __restrict__ grid, float* __restrict__ acc,
                       int B, int iters) {
  const int b = blockIdx.y;
  const float* __restrict__ gp = grid + (size_t)b * X3;
  const int tid    = blockIdx.x * blockDim.x + threadIdx.x;
  const int stride = gridDim.x * blockDim.x;  // in float4 units

  float tv = 0.f, ms = 0.f;
  for (int it = 0; it < iters; ++it) {
    int q = tid + it * stride;       // float4 index, exact coverage of X3/4
    int v = q << 2;                  // voxel linear index
    int xx = v & (XDIM - 1);
    int yy = (v >> 7) & (XDIM - 1);
    int zz = v >> 14;

    const float4 c = *(const float4*)(gp + v);

    // +x direction: 3 in-register diffs + boundary-masked 4th
    float a1 = c.y - c.x, a2 = c.z - c.y, a3 = c.w - c.z;
    float a4 = 0.f;
    if (xx < XDIM - 4) a4 = gp[v + 4] - c.w;
    tv += fabsf(a1) + fabsf(a2) + fabsf(a3) + fabsf(a4);
    ms += a1 * a1 + a2 * a2 + a3 * a3 + a4 * a4;

    if (yy < XDIM - 1) {
      const float4 n = *(const float4*)(gp + v + XDIM);
      float d0 = n.x - c.x, d1 = n.y - c.y, d2 = n.z - c.z, d3 = n.w - c.w;
      tv += fabsf(d0) + fabsf(d1) + fabsf(d2) + fabsf(d3);
      ms += d0 * d0 + d1 * d1 + d2 * d2 + d3 * d3;
    }
    if (zz < XDIM - 1) {
      const float4 n = *(const float4*)(gp + v + XDIM * XDIM);
      float d0 = n.x - c.x, d1 = n.y - c.y, d2 = n.z - c.z, d3 = n.w - c.w;
      tv += fabsf(d0) + fabsf(d1) + fabsf(d2) + fabsf(d3);
      ms += d0 * d0 + d1 * d1 + d2 * d2 + d3 * d3;
    }
    // stream-ahead hint for next grid-stride tile -> global_prefetch_b8
    if (it + 1 < iters) __builtin_prefetch((const void*)(gp + v + stride * 4), 0, 1);
  }

  // EXEC is all-ones here (uniform trip count, no early exit): WMMA-legal.
  float tvW = wave_sum32(tv);
  float msW = wave_sum32(ms);

  __shared__ float sT[8];
  __shared__ float sM[8];
  int w = threadIdx.x >> 5, lane = threadIdx.x & 31;
  if (lane == 0) { sT[w] = tvW; sM[w] = msW; }
  __syncthreads();
  if (threadIdx.x == 0) {
    float t = 0.f, m = 0.f;
    int nw = blockDim.x >> 5;
    for (int k = 0; k < nw; ++k) { t += sT[k]; m += sM[k]; }
    atomicAdd(acc + b, t);          // tv accumulator
    atomicAdd(acc + B + b, m);      // mse accumulator
  }
}

// ---------------------------------------------------------------- finalize
__global__ void dsv_finalize_kernel(const float* __restrict__ acc,
                                    float* __restrict__ out, int B) {
  int i = threadIdx.x;
  const float tvScale  = 1.0f / (float)X3;                       // X^3
  const float mseScale = 1.0f / (float)(2 * XDIM * XDIM - 2 * XDIM); // 2X^2-2X
  if (i < B)          out[i] = acc[i] * tvScale;
  else if (i < 2 * B) out[i] = acc[i] * mseScale;
}

// ---------------------------------------------------------------- launcher
extern "C" void kernel_launch(void* const* d_in, const int* in_sizes, int n_in,
                              void* d_out, int out_size, void* d_ws, size_t ws_size,
                              hipStream_t stream) {
  const int*   indices = (const int*)d_in[0];
  const float* values  = (const float*)d_in[1];

  const int N = in_sizes[0] / 3;        // 500000
  const int B = in_sizes[1] / N;        // 16

  float* acc  = (float*)d_ws;                         // [2*B] partial sums
  float* grid = (float*)((char*)d_ws + 256);          // B * X^3 floats

  // 1) zero grid + accumulators (one float4 store per thread)
  size_t total4 = (size_t)B * X3 / 4;                 // 8,388,608
  unsigned zb = (unsigned)(total4 / 256);             // exact division
  dsv_zero_kernel<<<zb, 256, 0, stream>>>((float4*)grid, acc, 2 * B);

  // 2) scatter-add
  dsv_scatter_kernel<<<(N + 255) / 256, 256, 0, stream>>>(indices, values,
                                                          grid, N, B);

  // 3) finite-difference reduction (256 blocks/batch * 256 thr * 8 iters * 4 vox = X^3)
  const int blocksX = 256;
  const int iters   = (X3 / 4) / (blocksX * 256);     // = 8
  dsv_reduce_kernel<<<dim3(blocksX, B), 256, 0, stream>>>(grid, acc, B, iters);

  // 4) scale + pack [tv(0..B-1), mse(0..B-1)]
  dsv_finalize_kernel<<<1, 64, 0, stream>>>(acc, (float*)d_out, B);
}